// GraphTransformer_1202590843049
// MI455X (gfx1250) — compile-verified
//
#include <hip/hip_runtime.h>
#include <hip/hip_bf16.h>
#include <math.h>

// ---------------------------------------------------------------------------
// GraphTransformer (TransformerConv x3 + BN/ReLU) for gfx1250 (MI455X).
// GEMMs use V_WMMA_F32_16X16X4_F32 (fp32 matrix pipe, wave32).
// ---------------------------------------------------------------------------

typedef __attribute__((ext_vector_type(2))) float v2f;
typedef __attribute__((ext_vector_type(8))) float v8f;

#define NNODES 50000
#define NEDGES 800000
#define DIN    64
#define HC     128   // heads * channels = 8 * 16
#define NHEAD  8
#define CH     16

// ---------------------------------------------------------------------------
// WMMA fp32 GEMM: C[M x 128] = A[M x Kd] @ B[Kd x 128] (+ bias[128])
// One wave per 16-row strip; 8 accumulators (16x16 tiles) covering 128 cols.
// f32 16x16x4 VGPR layout (ISA 7.12.2): A/B: VGPR v, lane-half h -> K = v+2h;
// lane&15 = M (for A) / N (for B).  C/D: VGPR r, half h -> M = r+8h, N=lane&15.
// ---------------------------------------------------------------------------
__global__ __launch_bounds__(256)
void gemm128_wmma(const float* __restrict__ A, const float* __restrict__ B,
                  const float* __restrict__ bias, float* __restrict__ C,
                  int M, int Kd) {
    int wave = blockIdx.x * (blockDim.x >> 5) + (threadIdx.x >> 5);
    int m0   = wave * 16;
    if (m0 >= M) return;                 // wave-uniform: EXEC stays all-ones
    int lane = threadIdx.x & 31;
    int half = lane >> 4;
    int idx  = lane & 15;

    v8f acc[8];
#pragma unroll
    for (int t = 0; t < 8; ++t)
#pragma unroll
        for (int r = 0; r < 8; ++r) acc[t][r] = 0.0f;

    const float* Arow = A + (size_t)(m0 + idx) * Kd + 2 * half;
    for (int k0 = 0; k0 < Kd; k0 += 4) {
        v2f a;
        a.x = Arow[k0];
        a.y = Arow[k0 + 1];
        const float* Bp = B + (size_t)(k0 + 2 * half) * HC + idx;
#pragma unroll
        for (int t = 0; t < 8; ++t) {
            v2f b;
            b.x = Bp[t * 16];
            b.y = Bp[HC + t * 16];
            acc[t] = __builtin_amdgcn_wmma_f32_16x16x4_f32(
                false, a, false, b, (short)0, acc[t], false, false);
        }
    }
#pragma unroll
    for (int t = 0; t < 8; ++t) {
        float bb = bias ? bias[t * 16 + idx] : 0.0f;
        float* Cp = C + (size_t)(m0 + 8 * half) * HC + t * 16 + idx;
#pragma unroll
        for (int r = 0; r < 8; ++r)
            Cp[(size_t)r * HC] = acc[t][r] + bb;
    }
}

// ---------------------------------------------------------------------------
__device__ __forceinline__ void atomicMaxF(float* addr, float val) {
    unsigned int* ua = (unsigned int*)addr;
    unsigned int old = *ua;
    while (true) {
        if (__uint_as_float(old) >= val) break;
        unsigned int assumed = old;
        old = atomicCAS(ua, assumed, __float_as_uint(val));
        if (old == assumed) break;
    }
}

// per-layer init: segment-max = -inf, segment-sum = 0, bn stats = 0
__global__ __launch_bounds__(256)
void init_layer_kernel(float* __restrict__ m, float* __restrict__ s,
                       float* __restrict__ stats, int nh) {
    int i = blockIdx.x * blockDim.x + threadIdx.x;
    if (i < nh) { m[i] = -__builtin_inff(); s[i] = 0.0f; }
    if (i < 256) stats[i] = 0.0f;
}

// alpha[e,h] = dot(q[dst], k[src]+e) / sqrt(c); segment max over dst
__global__ __launch_bounds__(256)
void attn_alpha_kernel(const float* __restrict__ Q, const float* __restrict__ K,
                       const float* __restrict__ Ep, const int* __restrict__ src,
                       const int* __restrict__ dst, float* __restrict__ alpha,
                       float* __restrict__ mseg, int total) {
    int t = blockIdx.x * blockDim.x + threadIdx.x;
    if (t >= total) return;
    int e = t >> 3, h = t & 7;
    int sn = src[e], dn = dst[e];
    const float4* qp = (const float4*)(Q + (size_t)dn * HC + h * CH);
    const float4* kp = (const float4*)(K + (size_t)sn * HC + h * CH);
    const float4* ep = (const float4*)(Ep + (size_t)e * HC + h * CH);
    float acc = 0.0f;
#pragma unroll
    for (int j = 0; j < 4; ++j) {
        float4 q = qp[j], k = kp[j], x = ep[j];
        acc += q.x * (k.x + x.x) + q.y * (k.y + x.y) +
               q.z * (k.z + x.z) + q.w * (k.w + x.w);
    }
    float al = acc * 0.25f;  // 1/sqrt(16)
    alpha[t] = al;
    atomicMaxF(&mseg[(size_t)dn * NHEAD + h], al);
}

// p = exp(alpha - m[dst]); segment sum of p over dst
__global__ __launch_bounds__(256)
void attn_exp_kernel(const int* __restrict__ dst, float* __restrict__ alpha,
                     const float* __restrict__ mseg, float* __restrict__ sseg,
                     int total) {
    int t = blockIdx.x * blockDim.x + threadIdx.x;
    if (t >= total) return;
    int e = t >> 3, h = t & 7;
    int dn = dst[e];
    float p = expf(alpha[t] - mseg[(size_t)dn * NHEAD + h]);
    alpha[t] = p;
    atomicAdd(&sseg[(size_t)dn * NHEAD + h], p);
}

// out[dst] += (v[src] + e) * p / (s[dst] + 1e-16)
__global__ __launch_bounds__(256)
void attn_msg_kernel(const float* __restrict__ V, const float* __restrict__ Ep,
                     const int* __restrict__ src, const int* __restrict__ dst,
                     const float* __restrict__ alpha, const float* __restrict__ sseg,
                     float* __restrict__ out, int total) {
    int t = blockIdx.x * blockDim.x + threadIdx.x;
    if (t >= total) return;
    int e = t >> 3, h = t & 7;
    int sn = src[e], dn = dst[e];
    float a = alpha[t] / (sseg[(size_t)dn * NHEAD + h] + 1e-16f);
    const float4* vp = (const float4*)(V + (size_t)sn * HC + h * CH);
    const float4* ep = (const float4*)(Ep + (size_t)e * HC + h * CH);
    float* op = out + (size_t)dn * HC + h * CH;
#pragma unroll
    for (int j = 0; j < 4; ++j) {
        float4 v = vp[j], x = ep[j];
        atomicAdd(op + j * 4 + 0, (v.x + x.x) * a);
        atomicAdd(op + j * 4 + 1, (v.y + x.y) * a);
        atomicAdd(op + j * 4 + 2, (v.z + x.z) * a);
        atomicAdd(op + j * 4 + 3, (v.w + x.w) * a);
    }
}

// BN stats: stats[0:128]=sum, stats[128:256]=sumsq.  Stride is a multiple of
// 128 so each thread touches exactly one channel -> coalesced accumulation.
__global__ __launch_bounds__(256)
void bn_stats_kernel(const float* __restrict__ X, float* __restrict__ stats,
                     int nrows) {
    int tid = threadIdx.x;
    size_t total = (size_t)nrows * HC;
    size_t stride = (size_t)gridDim.x * 256;
    float s = 0.0f, sq = 0.0f;
    for (size_t i = (size_t)blockIdx.x * 256 + tid; i < total; i += stride) {
        float v = X[i];
        s += v;
        sq += v * v;
    }
    __shared__ float shs[256], shq[256];
    shs[tid] = s; shq[tid] = sq;
    __syncthreads();
    if (tid < 128) {
        atomicAdd(&stats[tid],        shs[tid] + shs[tid + 128]);
        atomicAdd(&stats[128 + tid],  shq[tid] + shq[tid + 128]);
    }
}

// y = relu(g * (x - mu) * rsqrt(var + 1e-5) + b)
__global__ __launch_bounds__(256)
void bn_relu_kernel(const float* __restrict__ X, const float* __restrict__ stats,
                    const float* __restrict__ g, const float* __restrict__ b,
                    float* __restrict__ Y, int nrows) {
    int i = blockIdx.x * blockDim.x + threadIdx.x;
    if (i >= nrows * HC) return;
    int c = i & (HC - 1);
    float invN = 1.0f / (float)nrows;
    float mu  = stats[c] * invN;
    float var = stats[128 + c] * invN - mu * mu;
    float y = g[c] * (X[i] - mu) * rsqrtf(var + 1e-5f) + b[c];
    Y[i] = fmaxf(y, 0.0f);
}

// ---------------------------------------------------------------------------
extern "C" void kernel_launch(void* const* d_in, const int* in_sizes, int n_in,
                              void* d_out, int out_size, void* d_ws, size_t ws_size,
                              hipStream_t stream) {
    (void)in_sizes; (void)n_in; (void)out_size; (void)ws_size;
    const int N = NNODES, E = NEDGES;

    const float* x  = (const float*)d_in[0];
    const int*   ei = (const int*)d_in[1];
    const float* ea = (const float*)d_in[2];
    const int* src = ei;
    const int* dst = ei + E;

    // workspace carve-up (floats)
    float* ws = (float*)d_ws;
    size_t off = 0;
    float* Qb    = ws + off; off += (size_t)N * HC;
    float* Kb    = ws + off; off += (size_t)N * HC;
    float* Vb    = ws + off; off += (size_t)N * HC;
    float* hA    = ws + off; off += (size_t)N * HC;   // post-BN features
    float* hmsg  = ws + off; off += (size_t)N * HC;   // pre-BN accumulation
    float* mseg  = ws + off; off += (size_t)N * NHEAD;
    float* sseg  = ws + off; off += (size_t)N * NHEAD;
    float* stats = ws + off; off += 256;
    float* alpha = ws + off; off += (size_t)E * NHEAD;
    float* Ep    = ws + off; off += (size_t)E * HC;

    const int eh_total = E * NHEAD;                    // 6.4M
    const int eh_blocks = (eh_total + 255) / 256;
    const int nh_total = N * NHEAD;
    const int nh_blocks = (nh_total + 255) / 256;
    const int nf_blocks = (N * HC + 255) / 256;

    for (int l = 0; l < 3; ++l) {
        int base = 3 + l * 9;
        const float* Wq = (const float*)d_in[base + 0];
        const float* bq = (const float*)d_in[base + 1];
        const float* Wk = (const float*)d_in[base + 2];
        const float* bk = (const float*)d_in[base + 3];
        const float* Wv = (const float*)d_in[base + 4];
        const float* bv = (const float*)d_in[base + 5];
        const float* We = (const float*)d_in[base + 6];
        const float* Ws = (const float*)d_in[base + 7];
        const float* bs = (const float*)d_in[base + 8];

        const float* hin = (l == 0) ? x : hA;
        const int Kd     = (l == 0) ? DIN : HC;
        float* hout      = (l == 2) ? (float*)d_out : hmsg;

        const int ntile_blocks = ((N / 16) + 7) / 8;
        const int etile_blocks = ((E / 16) + 7) / 8;

        gemm128_wmma<<<ntile_blocks, 256, 0, stream>>>(hin, Wq, bq, Qb, N, Kd);
        gemm128_wmma<<<ntile_blocks, 256, 0, stream>>>(hin, Wk, bk, Kb, N, Kd);
        gemm128_wmma<<<ntile_blocks, 256, 0, stream>>>(hin, Wv, bv, Vb, N, Kd);
        gemm128_wmma<<<ntile_blocks, 256, 0, stream>>>(hin, Ws, bs, hout, N, Kd);
        gemm128_wmma<<<etile_blocks, 256, 0, stream>>>(ea, We, nullptr, Ep, E, DIN);

        init_layer_kernel<<<nh_blocks, 256, 0, stream>>>(mseg, sseg, stats, nh_total);

        attn_alpha_kernel<<<eh_blocks, 256, 0, stream>>>(Qb, Kb, Ep, src, dst,
                                                         alpha, mseg, eh_total);
        attn_exp_kernel<<<eh_blocks, 256, 0, stream>>>(dst, alpha, mseg, sseg,
                                                       eh_total);
        attn_msg_kernel<<<eh_blocks, 256, 0, stream>>>(Vb, Ep, src, dst, alpha,
                                                       sseg, hout, eh_total);

        if (l < 2) {
            const float* g = (const float*)d_in[30 + 2 * l];
            const float* b = (const float*)d_in[31 + 2 * l];
            bn_stats_kernel<<<512, 256, 0, stream>>>(hout, stats, N);
            bn_relu_kernel<<<nf_blocks, 256, 0, stream>>>(hout, stats, g, b, hA, N);
        }
    }

    // second tuple output: edge_attr passthrough
    hipMemcpyAsync((float*)d_out + (size_t)N * HC, ea,
                   (size_t)E * DIN * sizeof(float),
                   hipMemcpyDeviceToDevice, stream);
}